// RNN_classifier_65979287601758
// MI455X (gfx1250) — compile-verified
//
#include <hip/hip_runtime.h>

// ---------------------------------------------------------------------------
// RNN scan + fused GEMV for MI455X (gfx1250).
//
// Roofline: mandatory reads = 5 arrays x 127 rows x 262144 cols x 4B ~= 666 MB
// -> ~29 us at 23.3 TB/s. Memory bound; the final x_bar @ fc_w GEMV is fused
// into the scan as a per-timestep wave reduction (x_bar never materialized,
// saving 268 MB of traffic). Inputs are streamed with CDNA5 async global->LDS
// copies (ASYNCcnt-tracked, no in-flight dest VGPRs), distance-3 pipeline.
// Cross-lane reduction is pure-VALU: DPP16 row_xmask butterfly + permlanex16.
// Gate nonlinearities use the CDNA5 hardware v_tanh_f32 when available.
// ---------------------------------------------------------------------------

#define L_STEPS 128
#define D_DIM   262144
#define WG      256
#define NBLK    (D_DIM / WG)   // 1024 workgroups
#define NWAVE   (WG / 32)      // 8 waves per workgroup

// --- transcendentals: prefer CDNA5 v_tanh_f32; fallback v_exp/v_rcp ---------
#if __has_builtin(__builtin_amdgcn_tanhf)
__device__ __forceinline__ float fast_tanh(float z) {
  return __builtin_amdgcn_tanhf(z);                 // 1x v_tanh_f32
}
__device__ __forceinline__ float fast_sigmoid(float z) {
  // sigmoid(z) = 0.5*tanh(z/2) + 0.5
  return __builtin_fmaf(0.5f, __builtin_amdgcn_tanhf(0.5f * z), 0.5f);
}
#else
__device__ __forceinline__ float fast_sigmoid(float z) {
  // 1 / (1 + 2^(-z*log2e))
  return __builtin_amdgcn_rcpf(1.0f + __builtin_amdgcn_exp2f(-1.4426950408889634f * z));
}
__device__ __forceinline__ float fast_tanh(float z) {
  // 2 / (1 + 2^(-2z*log2e)) - 1
  return __builtin_fmaf(
      2.0f, __builtin_amdgcn_rcpf(1.0f + __builtin_amdgcn_exp2f(-2.8853900817779268f * z)),
      -1.0f);
}
#endif

// --- pure-VALU wave32 sum: DPP row_xmask butterfly + permlanex16 ------------
template <int CTRL>
__device__ __forceinline__ float dpp_xadd(float v) {
  int s = __builtin_bit_cast(int, v);
  int p = __builtin_amdgcn_update_dpp(0, s, CTRL, 0xf, 0xf, true);
  return v + __builtin_bit_cast(float, p);
}
__device__ __forceinline__ float wave_sum32(float c) {
  c = dpp_xadd<0x161>(c);  // row_xmask:1
  c = dpp_xadd<0x162>(c);  // row_xmask:2
  c = dpp_xadd<0x164>(c);  // row_xmask:4
  c = dpp_xadd<0x168>(c);  // row_xmask:8
  int s = __builtin_bit_cast(int, c);
  int p = __builtin_amdgcn_permlanex16(s, s, 0x76543210, 0xfedcba98, true, false);
  return c + __builtin_bit_cast(float, p);   // all 32 lanes hold the total
}

// --- CDNA5 async global->LDS copy (ASYNCcnt-tracked, no dest VGPR) ----------
__device__ __forceinline__ void async_ld_b32(const float* saddr, unsigned voff_bytes,
                                             unsigned lds_byte_off) {
  // GVS mode: loads dword at saddr + vaddr byte offset into LDS at byte address vdst
  asm volatile("global_load_async_to_lds_b32 %0, %1, %2"
               :: "v"(lds_byte_off), "v"(voff_bytes), "s"(saddr)
               : "memory");
}
__device__ __forceinline__ void wait_async_le15() { asm volatile("s_wait_asynccnt 15" ::: "memory"); }
__device__ __forceinline__ void wait_async_le10() { asm volatile("s_wait_asynccnt 10" ::: "memory"); }
__device__ __forceinline__ void wait_async_le5()  { asm volatile("s_wait_asynccnt 5"  ::: "memory"); }
__device__ __forceinline__ void wait_async_le0()  { asm volatile("s_wait_asynccnt 0"  ::: "memory"); }
__device__ __forceinline__ void wait_ds_le0()     { asm volatile("s_wait_dscnt 0"     ::: "memory"); }

// ---------------------------------------------------------------------------
// Kernel 1: per-column scan, fused dot with fc_w, per-block partials to d_ws.
// partials layout: [t * NBLK + block]  (contiguous per t for kernel 2)
// ---------------------------------------------------------------------------
__global__ __launch_bounds__(WG) void rnn_scan_kernel(
    const float* __restrict__ x,   const float* __restrict__ w_x,
    const float* __restrict__ w_g, const float* __restrict__ w_u,
    const float* __restrict__ msk, const float* __restrict__ fcw,
    float* __restrict__ partials)
{
  // 4-deep ring of 5 input rows: 5 * 4 * 256 * 4B = 20 KB
  __shared__ float sx [4][WG];
  __shared__ float sm [4][WG];
  __shared__ float swx[4][WG];
  __shared__ float swu[4][WG];
  __shared__ float swg[4][WG];
  // per-wave partial sums per timestep (each written exactly once -> deterministic)
  __shared__ float wpart[L_STEPS][NWAVE];

  const int tid  = threadIdx.x;
  const int lane = tid & 31;
  const int wid  = tid >> 5;
  const unsigned col = blockIdx.x * WG + tid;

  // LDS byte offsets of this lane's slot in buffer 0 (flat->LDS: low 32 bits)
  const unsigned lo_x  = (unsigned)(size_t)&sx [0][tid];
  const unsigned lo_m  = (unsigned)(size_t)&sm [0][tid];
  const unsigned lo_wx = (unsigned)(size_t)&swx[0][tid];
  const unsigned lo_wu = (unsigned)(size_t)&swu[0][tid];
  const unsigned lo_wg = (unsigned)(size_t)&swg[0][tid];

  const float fw = fcw[col];

  // wave reduce; lane0 stores the per-wave sum (no atomics, deterministic)
  auto wred = [&](int j, float c) {
    c = wave_sum32(c);
    if (lane == 0) wpart[j][wid] = c;
  };

  // issue one prefetch stage (row t of all 5 arrays) into ring slot b
  auto issue = [&](int t, int b) {
    const unsigned voff = ((unsigned)t * (unsigned)D_DIM + col) * 4u;
    const unsigned bo   = (unsigned)b * (WG * 4u);
    async_ld_b32(x,   voff, lo_x  + bo);
    async_ld_b32(msk, voff, lo_m  + bo);
    async_ld_b32(w_x, voff, lo_wx + bo);
    async_ld_b32(w_u, voff, lo_wu + bo);
    async_ld_b32(w_g, voff, lo_wg + bo);
  };

  // ---- t = 0 prologue (row 0 of x, weight_x, weight_g; mask[0]/w_u[0] unused)
  const float x0  = x  [col];
  const float wx0 = w_x[col];
  const float wg0 = w_g[col];

  wred(0, x0 * fw);                       // x_bar[0] = x[0]

  const float h0 = fast_tanh(x0 * wx0);
  float h    = h0;
  float xbar = __builtin_fmaf(h0, wg0, x0);   // x_bar[1]
  wred(1, xbar * fw);

  // one scan step using ring slot b; contributes x_bar[j] = new xbar
  auto step = [&](int b, int j) {
    const float xt  = sx [b][tid];
    const float mt  = sm [b][tid];
    const float wxt = swx[b][tid];
    const float wut = swu[b][tid];
    const float wgt = swg[b][tid];
    const float x_trans = __builtin_fmaf(mt, xt - xbar, xbar);   // x*m + xbar*(1-m)
    const float u = fast_tanh(x_trans * wxt);
    const float f = fast_sigmoid(__builtin_fmaf(u, wut, h)) * mt;
    h = __builtin_fmaf(f, h - u, u);                              // f*h + (1-f)*u
    h = fmaxf(h, h0);                                             // h0 + relu(h-h0)
    xbar = __builtin_fmaf(h, wgt, x_trans);
    wred(j, xbar * fw);
  };

  // ---- software pipeline: distance-3 prefetch, 4-slot ring, in-order retire
  issue(1, 1);
  issue(2, 2);
  issue(3, 3);                 // 15 async loads in flight
  for (int t = 1; t <= 123; ++t) {
    wait_ds_le0();             // prior LDS reads retired before slot reuse
    issue(t + 3, (t + 3) & 3); // 20 in flight
    wait_async_le15();         // oldest stage (t) retired (in-order per wave)
    step(t & 3, t + 1);
  }
  wait_async_le10();           // stage 124 retired
  step(124 & 3, 125);
  wait_async_le5();            // stage 125 retired
  step(125 & 3, 126);
  wait_async_le0();            // stage 126 retired
  step(126 & 3, 127);

  // ---- deterministic block reduction: fixed-order sum over 8 waves
  __syncthreads();
  if (tid < L_STEPS) {
    float s = 0.0f;
#pragma unroll
    for (int w = 0; w < NWAVE; ++w) s += wpart[tid][w];
    partials[tid * NBLK + blockIdx.x] = s;
  }
}

// ---------------------------------------------------------------------------
// Kernel 2: reduce 1024 block partials per timestep, sigmoid(dot + fc_b).
// ---------------------------------------------------------------------------
__global__ __launch_bounds__(256) void finalize_kernel(
    const float* __restrict__ partials, const float* __restrict__ fcb,
    float* __restrict__ out)
{
  __shared__ float wsum[8];
  const int t   = blockIdx.x;
  const int tid = threadIdx.x;

  float s = 0.0f;
  for (int i = tid; i < NBLK; i += 256) s += partials[t * NBLK + i];
  s = wave_sum32(s);
  if ((tid & 31) == 0) wsum[tid >> 5] = s;
  __syncthreads();
  if (tid == 0) {
    float acc = 0.0f;
#pragma unroll
    for (int w = 0; w < 8; ++w) acc += wsum[w];
    out[t] = fast_sigmoid(acc + fcb[0]);
  }
}

// ---------------------------------------------------------------------------
extern "C" void kernel_launch(void* const* d_in, const int* in_sizes, int n_in,
                              void* d_out, int out_size, void* d_ws, size_t ws_size,
                              hipStream_t stream) {
  (void)in_sizes; (void)n_in; (void)out_size; (void)ws_size;
  const float* x   = (const float*)d_in[0];
  const float* w_x = (const float*)d_in[1];
  const float* w_g = (const float*)d_in[2];
  const float* w_u = (const float*)d_in[3];
  const float* msk = (const float*)d_in[4];
  const float* fcw = (const float*)d_in[5];
  const float* fcb = (const float*)d_in[6];
  float* out       = (float*)d_out;
  float* partials  = (float*)d_ws;   // needs L_STEPS * NBLK * 4 = 512 KB

  rnn_scan_kernel<<<NBLK, WG, 0, stream>>>(x, w_x, w_g, w_u, msk, fcw, partials);
  finalize_kernel<<<L_STEPS, 256, 0, stream>>>(partials, fcb, out);
}